// SpikeFP32RMSNormFullFP64_76312978916077
// MI455X (gfx1250) — compile-verified
//
#include <hip/hip_runtime.h>

// RMSNorm over last axis: x:(4,8192,2048) fp32, weight:(2048,) fp32 -> out same as x.
// Bandwidth-bound: 512 MB total traffic, ~22 us floor at 23.3 TB/s.
// Strategy: 1 block per row, row lives in registers, single HBM read + single HBM write,
// wave32 ds_swizzle reduction, non-temporal b128 loads/stores (streams > 192 MB L2).

typedef float v4f __attribute__((ext_vector_type(4)));

#define RMS_EPS 1e-6f

constexpr int NFEAT = 2048;             // features per row (last axis)
constexpr int BLOCK = 256;              // 8 wave32 waves per WGP-resident block
constexpr int WAVES = BLOCK / 32;
constexpr int V4_PER_THREAD = NFEAT / 4 / BLOCK;  // 2 float4 per thread

static_assert(NFEAT % (4 * BLOCK) == 0, "row must tile into float4s per thread");

__device__ __forceinline__ float wave32_reduce_add(float v) {
    // ds_swizzle_b32 group-of-32 mode: offset = (xor_mask<<10) | (or_mask<<5) | and_mask
    // XOR butterfly across all 32 lanes of the wave.
    int b = __builtin_bit_cast(int, v);
    v += __builtin_bit_cast(float, __builtin_amdgcn_ds_swizzle(b, (1  << 10) | 0x1f));
    b = __builtin_bit_cast(int, v);
    v += __builtin_bit_cast(float, __builtin_amdgcn_ds_swizzle(b, (2  << 10) | 0x1f));
    b = __builtin_bit_cast(int, v);
    v += __builtin_bit_cast(float, __builtin_amdgcn_ds_swizzle(b, (4  << 10) | 0x1f));
    b = __builtin_bit_cast(int, v);
    v += __builtin_bit_cast(float, __builtin_amdgcn_ds_swizzle(b, (8  << 10) | 0x1f));
    b = __builtin_bit_cast(int, v);
    v += __builtin_bit_cast(float, __builtin_amdgcn_ds_swizzle(b, (16 << 10) | 0x1f));
    return v;
}

__global__ void __launch_bounds__(BLOCK)
rmsnorm_f32_kernel(const float* __restrict__ x,
                   const float* __restrict__ w,
                   float* __restrict__ out) {
    const size_t row = blockIdx.x;
    const v4f* __restrict__ xr = reinterpret_cast<const v4f*>(x + row * (size_t)NFEAT);
    v4f* __restrict__ orow     = reinterpret_cast<v4f*>(out + row * (size_t)NFEAT);
    const v4f* __restrict__ w4 = reinterpret_cast<const v4f*>(w);

    const int t = threadIdx.x;

    // ---- single streaming read of the row into registers (global_load_b128, TH=NT) ----
    v4f a = __builtin_nontemporal_load(&xr[t]);
    v4f c = __builtin_nontemporal_load(&xr[t + BLOCK]);

    // ---- sum of squares: 8 fp32 FMAs per thread ----
    float ss = a.x * a.x;
    ss = fmaf(a.y, a.y, ss);
    ss = fmaf(a.z, a.z, ss);
    ss = fmaf(a.w, a.w, ss);
    ss = fmaf(c.x, c.x, ss);
    ss = fmaf(c.y, c.y, ss);
    ss = fmaf(c.z, c.z, ss);
    ss = fmaf(c.w, c.w, ss);

    // ---- intra-wave reduction (wave32 ds_swizzle butterfly) ----
    ss = wave32_reduce_add(ss);

    // ---- cross-wave reduction through LDS (8 partials) ----
    __shared__ float part[WAVES];
    if ((t & 31) == 0) part[t >> 5] = ss;
    __syncthreads();

    float total = 0.0f;
#pragma unroll
    for (int i = 0; i < WAVES; ++i) total += part[i];  // LDS broadcast reads, conflict-free

    // ---- inv_rms = rsqrt(mean_sq + eps), hardware rsqrt + 1 Newton step ----
    const float m = fmaf(total, 1.0f / (float)NFEAT, RMS_EPS);
    float r = rsqrtf(m);
    r = r * fmaf(-0.5f * m * r, r, 1.5f);

    // ---- scale by inv_rms and weight; weight stays hot in L2 (regular TH) ----
    const v4f wa = w4[t];
    const v4f wc = w4[t + BLOCK];
    v4f oa = a * r * wa;
    v4f oc = c * r * wc;

    // ---- single streaming write (global_store_b128, TH=NT: no reuse, > L2 capacity) ----
    __builtin_nontemporal_store(oa, &orow[t]);
    __builtin_nontemporal_store(oc, &orow[t + BLOCK]);
}

extern "C" void kernel_launch(void* const* d_in, const int* in_sizes, int n_in,
                              void* d_out, int out_size, void* d_ws, size_t ws_size,
                              hipStream_t stream) {
    const float* x = (const float*)d_in[0];
    const float* w = (const float*)d_in[1];
    float* out = (float*)d_out;

    const int rows = in_sizes[0] / NFEAT;  // 4*8192 = 32768 rows
    rmsnorm_f32_kernel<<<dim3(rows), dim3(BLOCK), 0, stream>>>(x, w, out);
}